// RetinaNet_46420006535871
// MI455X (gfx1250) — compile-verified
//
#include <hip/hip_runtime.h>
#include <stdint.h>

typedef unsigned long long u64;
typedef unsigned int u32;

#define B_IMG   8
#define N_ANCH  196416
#define TOPK_K  1000
#define NPRED   100
#define SCALE_CLAMP_F 4.135166556742356f
#define TOTAL_VEC_THREADS 392832   // B * 9 * (21824/4)

// ---- float <-> order-preserving sortable key -------------------------------
__device__ __forceinline__ u32 f2key(float f) {
    u32 u = __float_as_uint(f);
    return (u & 0x80000000u) ? ~u : (u | 0x80000000u);
}
__device__ __forceinline__ float key2f(u32 k) {
    return __uint_as_float((k & 0x80000000u) ? (k ^ 0x80000000u) : ~k);
}

// ============================================================================
// Kernel A: streaming max/argmax over 80 classes; 503 MB read => HBM-bound
// (~22 us floor at 23.3 TB/s). Each thread handles 4 consecutive spatial
// positions (float4 = b128 loads; 32 lanes x 16B = 512B contiguous per wave
// per channel plane). Channel loop unrolled 8x for in-wave MLP across the
// 64KB-strided planes; global_prefetch_b8 reaches 8 planes ahead.
// Emits one 64-bit key per anchor: [scorekey:32][(262143-idx):18][cat:7][0:7]
// ============================================================================
__global__ void score_argmax_kernel(
    const float* __restrict__ lg0, const float* __restrict__ lg1,
    const float* __restrict__ lg2, const float* __restrict__ lg3,
    const float* __restrict__ lg4, u64* __restrict__ keys)
{
    int T = blockIdx.x * blockDim.x + threadIdx.x;
    if (T >= TOTAL_VEC_THREADS) return;

    // level decomposition (vec4 thread counts: 294912,73728,18432,4608,1152)
    int rem = T, l;
    if (rem < 294912) { l = 0; }
    else { rem -= 294912;
      if (rem < 73728) { l = 1; }
      else { rem -= 73728;
        if (rem < 18432) { l = 2; }
        else { rem -= 18432;
          if (rem < 4608) { l = 3; }
          else { rem -= 4608; l = 4; } } } }

    const int sL = (l==0)?16384:(l==1)?4096:(l==2)?1024:(l==3)?256:64;
    const int aB = (l==0)?0:(l==1)?147456:(l==2)?184320:(l==3)?193536:195840;
    const float* lg = (l==0)?lg0:(l==1)?lg1:(l==2)?lg2:(l==3)?lg3:lg4;

    const int sv      = sL >> 2;
    const int per_img = 9 * sv;
    const int b  = rem / per_img;  rem -= b * per_img;
    const int a  = rem / sv;
    const int p0 = (rem - a * sv) * 4;

    const float* basep = lg + ((size_t)(b * 720 + a * 80)) * (size_t)sL + p0;

    float4 m;
    m.x = m.y = m.z = m.w = -3.402823466e+38f;
    int amx = 0, amy = 0, amz = 0, amw = 0;

    #pragma unroll 8
    for (int c = 0; c < 80; ++c) {
        // one prefetch per unroll-8 batch, 8 planes ahead (compile-time cond)
        if ((c & 7) == 0 && c + 8 < 80)
            __builtin_prefetch(basep + (size_t)(c + 8) * sL, 0, 0);
        float4 v = *reinterpret_cast<const float4*>(basep + (size_t)c * sL);
        if (v.x > m.x) { m.x = v.x; amx = c; }
        if (v.y > m.y) { m.y = v.y; amy = c; }
        if (v.z > m.z) { m.z = v.z; amz = c; }
        if (v.w > m.w) { m.w = v.w; amw = c; }
    }

    float mv[4] = { m.x, m.y, m.z, m.w };
    int   av[4] = { amx, amy, amz, amw };
    u64* kout = keys + (size_t)b * N_ANCH;
    #pragma unroll
    for (int j = 0; j < 4; ++j) {
        // sigmoid(max logit) == max(sigmoid(logits)); valid <=> prob >= 0.05
        float sc    = 1.0f / (1.0f + expf(-mv[j]));
        float score = (sc >= 0.05f) ? sc : -1.0f;
        int gidx = aB + (p0 + j) * 9 + a;
        u64 k64 = ((u64)f2key(score) << 32)
                | ((u64)(u32)((262143 - gidx) & 0x3FFFF) << 14)
                | ((u64)(u32)av[j] << 7);
        kout[gidx] = k64;
    }
}

// ============================================================================
// Kernel B: exact top-1000 per image. One 1024-thread WG per image.
// 8-pass byte radix-select on distinct 64-bit keys -> exact 1000th key,
// LDS compact, 1024-wide bitonic sort (descending), write sorted keys.
// ============================================================================
__global__ void __launch_bounds__(1024) topk_select_kernel(
    const u64* __restrict__ keys, u64* __restrict__ sel)
{
    const int b   = blockIdx.x;
    const int tid = threadIdx.x;
    const u64* kp = keys + (size_t)b * N_ANCH;

    __shared__ u32 hist[256];
    __shared__ u64 selsh[1024];
    __shared__ u64 sh_prefix;
    __shared__ int sh_rem;
    __shared__ u32 sh_cnt;

    u64 prefix = 0, mask = 0;
    int rem = TOPK_K;

    for (int pass = 7; pass >= 0; --pass) {
        const int shift = pass * 8;
        if (tid < 256) hist[tid] = 0;
        __syncthreads();
        for (int i = tid; i < N_ANCH; i += 1024) {
            u64 kk = kp[i];
            if ((kk & mask) == prefix)
                atomicAdd(&hist[(u32)((kk >> shift) & 0xFFull)], 1u);
        }
        __syncthreads();
        if (tid == 0) {
            int r = rem, chosen = 0;
            for (int bb = 255; bb >= 0; --bb) {
                int c = (int)hist[bb];
                if (r > c) r -= c;
                else { chosen = bb; break; }
            }
            sh_rem    = r;
            sh_prefix = prefix | ((u64)(u32)chosen << shift);
        }
        __syncthreads();
        prefix = sh_prefix;
        rem    = sh_rem;
        mask  |= (0xFFull << shift);
    }
    // prefix is now the exact 1000th-largest key; select key >= prefix
    if (tid == 0) sh_cnt = 0;
    __syncthreads();
    for (int i = tid; i < N_ANCH; i += 1024) {
        u64 kk = kp[i];
        if (kk >= prefix) {
            u32 pos = atomicAdd(&sh_cnt, 1u);
            if (pos < 1024) selsh[pos] = kk;
        }
    }
    __syncthreads();
    u32 cnt = sh_cnt;
    if (tid >= (int)cnt) selsh[tid] = 0ull;   // pad (real keys are > 0)
    __syncthreads();

    // bitonic sort, 1024 elements, descending
    for (u32 ksz = 2; ksz <= 1024; ksz <<= 1) {
        for (u32 j = ksz >> 1; j > 0; j >>= 1) {
            u32 i = (u32)tid, ixj = i ^ j;
            if (ixj > i) {
                u64 x = selsh[i], y = selsh[ixj];
                bool asc  = ((i & ksz) != 0);
                bool swap = asc ? (x > y) : (x < y);
                if (swap) { selsh[i] = y; selsh[ixj] = x; }
            }
            __syncthreads();
        }
    }
    sel[(size_t)b * 1024 + tid] = selsh[tid];
}

// ============================================================================
// Kernel C: decode + class-aware greedy NMS + emit top-100. One WG per image.
// Selected keys staged into LDS via CDNA5 async global->LDS + s_wait_asynccnt.
// ============================================================================
__global__ void __launch_bounds__(1024) decode_nms_kernel(
    const u64* __restrict__ sel,
    const float* __restrict__ rg0, const float* __restrict__ rg1,
    const float* __restrict__ rg2, const float* __restrict__ rg3,
    const float* __restrict__ rg4,
    const float* __restrict__ anch,
    float* __restrict__ out)
{
    const int b   = blockIdx.x;
    const int tid = threadIdx.x;

    __shared__ u64   selsh[1024];
    __shared__ float sx1[1024], sy1[1024], sx2[1024], sy2[1024], areaL[1024];
    __shared__ int   keepL[1024];
    __shared__ int   rankL[1024];
    __shared__ float outbuf[NPRED * 6];

    // ---- CDNA5 async copy: 8 KB of sorted keys, global -> LDS --------------
    {
        // low 32 bits of a generic pointer to LDS == the LDS byte offset
        unsigned ldsoff = (unsigned)(uintptr_t)&selsh[tid];
        u64 gaddr = (u64)(uintptr_t)(sel + (size_t)b * 1024 + tid);
        asm volatile("global_load_async_to_lds_b64 %0, %1, off"
                     :: "v"(ldsoff), "v"(gaddr) : "memory");
        asm volatile("s_wait_asynccnt 0" ::: "memory");
    }
    __syncthreads();

    float score = -1.0f;
    int   cat   = 0;
    float bx1 = 0.f, by1 = 0.f, bx2 = 0.f, by2 = 0.f;
    int   kflag = 0;

    if (tid < TOPK_K) {
        u64 kk = selsh[tid];
        score  = key2f((u32)(kk >> 32));
        int idx = 262143 - (int)((kk >> 14) & 0x3FFFFull);
        cat     = (int)((kk >> 7) & 0x7Full);

        int l, li;
        if      (idx < 147456) { l = 0; li = idx; }
        else if (idx < 184320) { l = 1; li = idx - 147456; }
        else if (idx < 193536) { l = 2; li = idx - 184320; }
        else if (idx < 195840) { l = 3; li = idx - 193536; }
        else                   { l = 4; li = idx - 195840; }

        const int sL = (l==0)?16384:(l==1)?4096:(l==2)?1024:(l==3)?256:64;
        const float* rg = (l==0)?rg0:(l==1)?rg1:(l==2)?rg2:(l==3)?rg3:rg4;
        int p = li / 9;
        int a = li - p * 9;

        size_t off = ((size_t)(b * 36 + a * 4)) * (size_t)sL + p;
        float d0 = rg[off];
        float d1 = rg[off + (size_t)sL];
        float d2 = rg[off + 2 * (size_t)sL];
        float d3 = rg[off + 3 * (size_t)sL];

        float ax1 = anch[(size_t)idx * 4 + 0], ay1 = anch[(size_t)idx * 4 + 1];
        float ax2 = anch[(size_t)idx * 4 + 2], ay2 = anch[(size_t)idx * 4 + 3];
        float w  = ax2 - ax1, h = ay2 - ay1;
        float cx = (ax1 + ax2) * 0.5f, cy = (ay1 + ay2) * 0.5f;
        float dw = fminf(d2, SCALE_CLAMP_F), dh = fminf(d3, SCALE_CLAMP_F);
        float pcx = d0 * w + cx, pcy = d1 * h + cy;
        float pw = expf(dw) * w, ph = expf(dh) * h;
        bx1 = fminf(fmaxf(pcx - 0.5f * pw, 0.0f), 1023.0f);
        by1 = fminf(fmaxf(pcy - 0.5f * ph, 0.0f), 1023.0f);
        bx2 = fminf(fmaxf(pcx + 0.5f * pw, 0.0f), 1023.0f);
        by2 = fminf(fmaxf(pcy + 0.5f * ph, 0.0f), 1023.0f);
        kflag = (score >= 0.0f) ? 1 : 0;
    }

    // class-offset boxes for class-aware NMS (reference: boxes + cat*1024)
    float shift = (float)cat * 1024.0f;
    float fx1 = bx1 + shift, fy1 = by1 + shift;
    float fx2 = bx2 + shift, fy2 = by2 + shift;
    sx1[tid] = fx1; sy1[tid] = fy1; sx2[tid] = fx2; sy2[tid] = fy2;
    areaL[tid] = (fx2 - fx1) * (fy2 - fy1);
    keepL[tid] = kflag;
    __syncthreads();

    // greedy NMS scan: surviving i suppresses j>i with IoU > 0.5
    for (int i = 0; i < TOPK_K; ++i) {
        if (keepL[i] && tid > i && tid < TOPK_K && keepL[tid]) {
            float lx = fmaxf(sx1[i], fx1);
            float ly = fmaxf(sy1[i], fy1);
            float rx = fminf(sx2[i], fx2);
            float ry = fminf(sy2[i], fy2);
            float iw = fmaxf(rx - lx, 0.0f);
            float ih = fmaxf(ry - ly, 0.0f);
            float inter = iw * ih;
            float iou = inter / (areaL[i] + areaL[tid] - inter + 1e-9f);
            if (iou > 0.5f) keepL[tid] = 0;
        }
        __syncthreads();
    }

    if (tid == 0) {
        int r = 0;
        for (int i = 0; i < TOPK_K; ++i)
            rankL[i] = keepL[i] ? (r++) : -1;
    }
    for (int i = tid; i < NPRED * 6; i += 1024) outbuf[i] = -1.0f;
    __syncthreads();

    if (tid < TOPK_K && keepL[tid]) {
        int r = rankL[tid];
        if (r >= 0 && r < NPRED) {
            outbuf[r * 6 + 0] = (float)cat;
            outbuf[r * 6 + 1] = score;
            outbuf[r * 6 + 2] = bx1;
            outbuf[r * 6 + 3] = by1;
            outbuf[r * 6 + 4] = bx2;
            outbuf[r * 6 + 5] = by2;
        }
    }
    __syncthreads();
    for (int i = tid; i < NPRED * 6; i += 1024)
        out[(size_t)b * NPRED * 6 + i] = outbuf[i];
}

// ============================================================================
// launch: d_in order = logits0,regress0,logits1,regress1,...,logits4,regress4,anchors
// ws layout: [ keys: 8*196416 u64 ][ sel: 8*1024 u64 ]  (~12.64 MB)
// ============================================================================
extern "C" void kernel_launch(void* const* d_in, const int* in_sizes, int n_in,
                              void* d_out, int out_size, void* d_ws, size_t ws_size,
                              hipStream_t stream)
{
    (void)in_sizes; (void)n_in; (void)out_size; (void)ws_size;

    const float* lg[5];
    const float* rg[5];
    for (int i = 0; i < 5; ++i) {
        lg[i] = (const float*)d_in[2 * i];
        rg[i] = (const float*)d_in[2 * i + 1];
    }
    const float* anch = (const float*)d_in[10];

    u64* keys = (u64*)d_ws;
    u64* sel  = keys + (size_t)B_IMG * N_ANCH;

    const int blkA = 256;
    const int grdA = (TOTAL_VEC_THREADS + blkA - 1) / blkA;

    score_argmax_kernel<<<grdA, blkA, 0, stream>>>(
        lg[0], lg[1], lg[2], lg[3], lg[4], keys);

    topk_select_kernel<<<B_IMG, 1024, 0, stream>>>(keys, sel);

    decode_nms_kernel<<<B_IMG, 1024, 0, stream>>>(
        sel, rg[0], rg[1], rg[2], rg[3], rg[4], anch, (float*)d_out);
}